// STANetSAModule_45964740001823
// MI455X (gfx1250) — compile-verified
//
#include <hip/hip_runtime.h>
#include <math.h>

// Problem constants (from reference setup_inputs)
#define B_   2
#define N_   8192
#define M_   2048
#define C_   64
#define KNN  16
#define GN_EPS 1e-5f

typedef __attribute__((ext_vector_type(2))) float v2f;
typedef __attribute__((ext_vector_type(8))) float v8f;

// ---------------------------------------------------------------------------
// Kernel 1: brute-force k-NN (top-16 smallest L2 on xyz[:, :3]).
// 256 threads / block = 32 queries x 8 sub-threads. Each sub-thread scans
// 1024 points (LDS-staged float4 chunks -> ds_load_b128) keeping a register
// top-16; the 8 sorted sub-lists are merged per query through LDS.
// All queries in a block share batch b (32 | 2048). 128 blocks = 1024 waves.
// ---------------------------------------------------------------------------
#define KNN_QPB 32                 // queries per block
#define KNN_SUB 8                  // sub-threads per query
#define KNN_CH  1024               // points staged per chunk (16 KB as float4)

__global__ void knn_kernel(const float* __restrict__ xyz,
                           const float* __restrict__ new_xyz,
                           int* __restrict__ idx_out) {
  // 32 KB: first 16 KB doubles as float4 staging during the scan phase,
  // whole buffer becomes the (dist,int) merge arrays afterwards.
  __shared__ __align__(16) char smem_raw[KNN_QPB * KNN_SUB * KNN * 8];
  float4* sp = (float4*)smem_raw;

  const int t  = threadIdx.x;
  const int ql = t >> 3;                         // local query 0..31
  const int s  = t & 7;                          // sub-thread 0..7
  const int q  = blockIdx.x * KNN_QPB + ql;      // global query
  const int b  = q / M_;
  const float* qp = new_xyz + (size_t)q * 6;
  const float qx = qp[0], qy = qp[1], qz = qp[2];

  float bestd[KNN];
  int   besti[KNN];
#pragma unroll
  for (int j = 0; j < KNN; ++j) { bestd[j] = 3.4e38f; besti[j] = 0; }

  for (int c0 = 0; c0 < N_; c0 += KNN_CH) {
    __syncthreads();
    for (int i = t; i < KNN_CH; i += 256) {
      const float* p = xyz + (size_t)(b * N_ + c0 + i) * 6;
      sp[i] = make_float4(p[0], p[1], p[2], 0.f);
    }
    // prefetch next chunk while grinding on this one (global_prefetch_b8)
    if (c0 + KNN_CH < N_) {
      const char* np = (const char*)(xyz + (size_t)(b * N_ + c0 + KNN_CH) * 6);
      __builtin_prefetch(np + t * 96, 0, 0);
    }
    __syncthreads();

    const int i0 = s * (KNN_CH / KNN_SUB);       // this sub-thread's 128-pt slice
    for (int i = i0; i < i0 + KNN_CH / KNN_SUB; ++i) {
      float4 p = sp[i];
      float dx = qx - p.x, dy = qy - p.y, dz = qz - p.z;
      float d2 = dx * dx + dy * dy + dz * dz;
      if (d2 < bestd[KNN - 1]) {                 // strict <: ties keep lower idx
        bestd[KNN - 1] = d2;
        besti[KNN - 1] = c0 + i;
#pragma unroll
        for (int j = KNN - 1; j > 0; --j) {
          if (bestd[j] < bestd[j - 1]) {
            float td = bestd[j]; bestd[j] = bestd[j - 1]; bestd[j - 1] = td;
            int   ti = besti[j]; besti[j] = besti[j - 1]; besti[j - 1] = ti;
          }
        }
      }
    }
  }

  // ---- dump sorted sub-lists to LDS (staging buffer is dead now) ----
  __syncthreads();
  float* sd = (float*)smem_raw;                                   // 16 KB
  int*   si = (int*)(smem_raw + KNN_QPB * KNN_SUB * KNN * 4);     // 16 KB
  const int base = (ql * KNN_SUB + s) * KNN;
#pragma unroll
  for (int j = 0; j < KNN; ++j) { sd[base + j] = bestd[j]; si[base + j] = besti[j]; }
  __syncthreads();

  // ---- 8-way merge of sorted lists (one thread per query) ----
  if (s == 0) {
    int head[KNN_SUB];
#pragma unroll
    for (int u = 0; u < KNN_SUB; ++u) head[u] = 0;
    for (int out = 0; out < KNN; ++out) {
      float bd = 3.4e38f; int bi = 0x7fffffff; int bs = 0;
#pragma unroll
      for (int u = 0; u < KNN_SUB; ++u) {
        int h = head[u];
        if (h < KNN) {
          float d  = sd[(ql * KNN_SUB + u) * KNN + h];
          int   ii = si[(ql * KNN_SUB + u) * KNN + h];
          if (d < bd || (d == bd && ii < bi)) { bd = d; bi = ii; bs = u; }
        }
      }
      head[bs]++;
      idx_out[q * KNN + out] = bi;
    }
  }
}

// ---------------------------------------------------------------------------
// Kernel 2: per-group RANet + neighbor max-pool + concat.
// One wave32 per group. Lanes 0..15 own the 16 spatial bins / neighbors
// (lanes 16..31 mirror); GN reductions via shfl widths 16 (GN1) and 8 (GN2).
// Output h0[q][128] = concat(pooled_feat[64], ra_feat[64]).
// ---------------------------------------------------------------------------
__global__ void group_ranet_kernel(const float* __restrict__ xyz,
                                   const float* __restrict__ feat,
                                   const int*   __restrict__ idx,
                                   const float* __restrict__ c1w, const float* __restrict__ c1b,
                                   const float* __restrict__ g1w, const float* __restrict__ g1b,
                                   const float* __restrict__ c2w, const float* __restrict__ c2b,
                                   const float* __restrict__ g2w, const float* __restrict__ g2b,
                                   float* __restrict__ h0) {
  __shared__ float smem[8 * 512];                       // 8 waves * 32ch * 16pos
  const int lane = threadIdx.x & 31;
  const int wave = threadIdx.x >> 5;
  const int q = blockIdx.x * (blockDim.x >> 5) + wave;  // group id, 0..B*M-1
  const int b = q / M_;
  const int j = lane & 15;                              // spatial bin / neighbor id

  // ---- ramap: load neighbor j, compute range/azimuth, bin it ----
  const int nj = idx[q * KNN + j];
  const float* p = xyz + (size_t)(b * N_ + nj) * 6;
  const float px = p[0], py = p[1], rcs = p[3], vr = p[5];
  float rng = sqrtf(px * px + py * py);
  float az  = atan2f(py, px);
  float r_lo = rng, r_hi = rng, a_lo = az, a_hi = az;
  for (int off = 8; off; off >>= 1) {
    r_lo = fminf(r_lo, __shfl_xor(r_lo, off, 16));
    r_hi = fmaxf(r_hi, __shfl_xor(r_hi, off, 16));
    a_lo = fminf(a_lo, __shfl_xor(a_lo, off, 16));
    a_hi = fmaxf(a_hi, __shfl_xor(a_hi, off, 16));
  }
  float ur = (r_hi - r_lo) * 0.25f; if (ur == 0.f) ur = 1.f;
  float ua = (a_hi - a_lo) * 0.25f; if (ua == 0.f) ua = 1.f;
  int ri = (int)floorf((rng - r_lo) / ur); ri = min(max(ri, 0), 3);
  int ai = (int)floorf((az  - a_lo) / ua); ai = min(max(ai, 0), 3);
  const int bin = ri * 4 + ai;

  // lane j accumulates bin j (scatter-add / scatter-max starting from 0)
  float cnt = 0.f, rmax = 0.f, vmax = 0.f;
  for (int i = 0; i < 16; ++i) {
    int   bi = __shfl(bin, i, 16);
    float rc = __shfl(rcs, i, 16);
    float vv = __shfl(vr,  i, 16);
    if (bi == j) { cnt += 1.f; rmax = fmaxf(rmax, rc); vmax = fmaxf(vmax, vv); }
  }

  // ---- conv1 (1x1, 3->32) + bias ----
  float h1[32];
#pragma unroll
  for (int c = 0; c < 32; ++c)
    h1[c] = c1w[c * 3 + 0] * cnt + c1w[c * 3 + 1] * rmax + c1w[c * 3 + 2] * vmax + c1b[c];

  // ---- GN(8) over [4 ch x 16 pos] + ReLU ----
#pragma unroll
  for (int g = 0; g < 8; ++g) {
    float s = 0.f, s2 = 0.f;
#pragma unroll
    for (int c = 0; c < 4; ++c) { float v = h1[g * 4 + c]; s += v; s2 += v * v; }
    for (int off = 8; off; off >>= 1) {
      s  += __shfl_xor(s,  off, 16);
      s2 += __shfl_xor(s2, off, 16);
    }
    float mean = s * (1.f / 64.f);
    float var  = s2 * (1.f / 64.f) - mean * mean;
    float inv  = rsqrtf(var + GN_EPS);
#pragma unroll
    for (int c = 0; c < 4; ++c) {
      int ch = g * 4 + c;
      h1[ch] = fmaxf((h1[ch] - mean) * inv * g1w[ch] + g1b[ch], 0.f);
    }
  }

  // ---- exchange h1 through LDS for the 4x4 full-reduce conv2 ----
  float* wl = smem + wave * 512;
  if (lane < 16) {
#pragma unroll
    for (int c = 0; c < 32; ++c) wl[c * 16 + j] = h1[c];
  }
  __syncthreads();

  // ---- conv2 (4x4 VALID, 32->64): lane computes out ch lane & lane+32 ----
  float h2a = c2b[lane], h2b = c2b[lane + 32];
  for (int c = 0; c < 32; ++c) {
    const float* wa = c2w + ((size_t)(lane      * 32 + c)) * 16;
    const float* wb = c2w + ((size_t)((lane+32) * 32 + c)) * 16;
#pragma unroll
    for (int jj = 0; jj < 16; ++jj) {
      float v = wl[c * 16 + jj];
      h2a = fmaf(wa[jj], v, h2a);
      h2b = fmaf(wb[jj], v, h2b);
    }
  }

  // ---- GN(8) over 8 channels + ReLU (groups align to 8-lane segments) ----
  float sa = h2a, sa2 = h2a * h2a, sb = h2b, sb2 = h2b * h2b;
  for (int off = 4; off; off >>= 1) {
    sa += __shfl_xor(sa, off, 8); sa2 += __shfl_xor(sa2, off, 8);
    sb += __shfl_xor(sb, off, 8); sb2 += __shfl_xor(sb2, off, 8);
  }
  float ma = sa * 0.125f, va = sa2 * 0.125f - ma * ma;
  float mb = sb * 0.125f, vb = sb2 * 0.125f - mb * mb;
  h2a = fmaxf((h2a - ma) * rsqrtf(va + GN_EPS) * g2w[lane]      + g2b[lane],      0.f);
  h2b = fmaxf((h2b - mb) * rsqrtf(vb + GN_EPS) * g2w[lane + 32] + g2b[lane + 32], 0.f);

  // ---- neighbor feature max-pool (64 ch across 32 lanes x2) ----
  float pa = -3.4e38f, pb = -3.4e38f;
  for (int i = 0; i < KNN; ++i) {
    int ni = idx[q * KNN + i];
    const float* f = feat + (size_t)(b * N_ + ni) * C_;
    pa = fmaxf(pa, f[lane]);
    pb = fmaxf(pb, f[lane + 32]);
  }

  float* out = h0 + (size_t)q * 128;
  out[lane]      = pa;
  out[lane + 32] = pb;
  out[64 + lane] = h2a;
  out[96 + lane] = h2b;
}

// ---------------------------------------------------------------------------
// Kernel 3: 128x128 GEMM head via V_WMMA_F32_16X16X4_F32.
// Y[m][o] = sum_c X[m][c] * W[o][c] + bias[o].  One wave per 16x16 tile,
// K=128 -> 32 wmma issues.  A: 16x4 (lane&15=row, lane>>4 selects K pair);
// B: 4x16 = W^T tile, same per-lane addressing because B[k][n] = W[n][k].
// ---------------------------------------------------------------------------
__global__ void mlp_gemm_wmma_kernel(const float* __restrict__ X,   // [BM][128]
                                     const float* __restrict__ W,   // [128][128]
                                     const float* __restrict__ bias,
                                     float* __restrict__ Y) {       // [BM][128]
  const int lane = threadIdx.x & 31;
  const int wave = threadIdx.x >> 5;
  const int tile = blockIdx.x * (blockDim.x >> 5) + wave;  // 0 .. (BM/16)*8-1
  const int mt = tile >> 3;
  const int ot = tile & 7;
  const int m0 = mt * 16, o0 = ot * 16;

  const int r  = lane & 15;
  const int kk = (lane >> 4) << 1;                         // 0 or 2
  const float* xrow = X + (size_t)(m0 + r) * 128 + kk;
  const float* wrow = W + (size_t)(o0 + r) * 128 + kk;

  v8f acc = {};
  for (int k = 0; k < 128; k += 4) {
    v2f a = *(const v2f*)(xrow + k);                       // A: X[m0+r][k+kk .. +1]
    v2f bb = *(const v2f*)(wrow + k);                      // B: W[o0+r][k+kk .. +1]
    acc = __builtin_amdgcn_wmma_f32_16x16x4_f32(
        /*neg_a=*/false, a, /*neg_b=*/false, bb,
        /*c_mod=*/(short)0, acc, /*reuse_a=*/false, /*reuse_b=*/false);
  }

  const int o  = o0 + (lane & 15);
  const int hi = lane >> 4;
  const float bv = bias[o];
#pragma unroll
  for (int v = 0; v < 8; ++v) {
    int m = m0 + v + hi * 8;                               // C/D layout: VGPR v -> row
    Y[(size_t)m * 128 + o] = acc[v] + bv;
  }
}

// ---------------------------------------------------------------------------
// Kernel 4: GroupNorm stats over (b, group): 16 ch x M elements.
// One block per (b,g) -> 16 blocks.
// ---------------------------------------------------------------------------
__global__ void gn_stats_kernel(const float* __restrict__ X,  // [B*M][128]
                                float* __restrict__ stats) {  // [16][2]
  const int b = blockIdx.x >> 3, g = blockIdx.x & 7;
  const int total = M_ * 16;
  float s = 0.f, s2 = 0.f;
  for (int i = threadIdx.x; i < total; i += blockDim.x) {
    int m = i >> 4, c = (g << 4) + (i & 15);
    float v = X[((size_t)(b * M_ + m)) * 128 + c];
    s += v; s2 += v * v;
  }
  for (int off = 16; off; off >>= 1) {
    s  += __shfl_xor(s,  off, 32);
    s2 += __shfl_xor(s2, off, 32);
  }
  __shared__ float rs[16], rs2[16];
  const int wv = threadIdx.x >> 5;
  if ((threadIdx.x & 31) == 0) { rs[wv] = s; rs2[wv] = s2; }
  __syncthreads();
  if (threadIdx.x == 0) {
    const int nw = blockDim.x >> 5;
    float S = 0.f, S2 = 0.f;
    for (int i = 0; i < nw; ++i) { S += rs[i]; S2 += rs2[i]; }
    float mean = S / (float)total;
    float var  = S2 / (float)total - mean * mean;
    stats[blockIdx.x * 2]     = mean;
    stats[blockIdx.x * 2 + 1] = var;
  }
}

// ---------------------------------------------------------------------------
// Kernel 5: GroupNorm apply + ReLU (optionally writing [B,128,M] transposed).
// ---------------------------------------------------------------------------
__global__ void gn_apply_kernel(const float* __restrict__ X,
                                const float* __restrict__ stats,
                                const float* __restrict__ gw,
                                const float* __restrict__ gb,
                                float* __restrict__ Y,
                                int transpose_out) {
  const int i = blockIdx.x * blockDim.x + threadIdx.x;     // B*M*128 total
  const int c = i & 127;
  const int bm = i >> 7;
  const int b = bm / M_;
  const int m = bm - b * M_;
  const int g = c >> 4;
  const float mean = stats[(b * 8 + g) * 2];
  const float var  = stats[(b * 8 + g) * 2 + 1];
  float v = (X[i] - mean) * rsqrtf(var + GN_EPS) * gw[c] + gb[c];
  v = fmaxf(v, 0.f);
  if (transpose_out) Y[((size_t)(b * 128 + c)) * M_ + m] = v;
  else               Y[i] = v;
}

// ---------------------------------------------------------------------------
extern "C" void kernel_launch(void* const* d_in, const int* in_sizes, int n_in,
                              void* d_out, int out_size, void* d_ws, size_t ws_size,
                              hipStream_t stream) {
  const float* xyz     = (const float*)d_in[0];
  const float* new_xyz = (const float*)d_in[1];
  const float* feat    = (const float*)d_in[2];
  const float* c1w = (const float*)d_in[3];
  const float* c1b = (const float*)d_in[4];
  const float* g1w = (const float*)d_in[5];
  const float* g1b = (const float*)d_in[6];
  const float* c2w = (const float*)d_in[7];
  const float* c2b = (const float*)d_in[8];
  const float* g2w = (const float*)d_in[9];
  const float* g2b = (const float*)d_in[10];
  const float* m1w = (const float*)d_in[11];
  const float* m1b = (const float*)d_in[12];
  const float* n1w = (const float*)d_in[13];
  const float* n1b = (const float*)d_in[14];
  const float* m2w = (const float*)d_in[15];
  const float* m2b = (const float*)d_in[16];
  const float* n2w = (const float*)d_in[17];
  const float* n2b = (const float*)d_in[18];

  char* ws = (char*)d_ws;
  const size_t BM = (size_t)B_ * M_;
  int*   idx   = (int*)ws;                                   // BM*16 ints
  float* h0    = (float*)(ws + BM * KNN * 4);                // BM*128 f32
  float* t1    = (float*)(ws + BM * KNN * 4 + BM * 128 * 4); // BM*128 f32
  float* t2    = (float*)(ws + BM * KNN * 4 + 2 * BM * 128 * 4);
  float* stats = (float*)(ws + BM * KNN * 4 + 3 * BM * 128 * 4);

  // 1) k-NN indices (8-way split scan + merge: 1024 waves)
  knn_kernel<<<(B_ * M_) / KNN_QPB, 256, 0, stream>>>(xyz, new_xyz, idx);

  // 2) RANet + max-pool + concat -> h0 [BM][128]
  group_ranet_kernel<<<(B_ * M_) / 8, 256, 0, stream>>>(
      xyz, feat, idx, c1w, c1b, g1w, g1b, c2w, c2b, g2w, g2b, h0);

  // 3) mlp1: WMMA GEMM + bias -> t1; GN(8) over M + ReLU (in place)
  const int gemm_blocks = (int)((BM / 16) * 8 / 8);          // 8 waves / block
  mlp_gemm_wmma_kernel<<<gemm_blocks, 256, 0, stream>>>(h0, m1w, m1b, t1);
  gn_stats_kernel<<<16, 256, 0, stream>>>(t1, stats);
  gn_apply_kernel<<<(int)(BM * 128 / 256), 256, 0, stream>>>(t1, stats, n1w, n1b, t1, 0);

  // 4) mlp2: WMMA GEMM + bias -> t2; GN(8) + ReLU -> d_out as [B,128,M]
  mlp_gemm_wmma_kernel<<<gemm_blocks, 256, 0, stream>>>(t1, m2w, m2b, t2);
  gn_stats_kernel<<<16, 256, 0, stream>>>(t2, stats);
  gn_apply_kernel<<<(int)(BM * 128 / 256), 256, 0, stream>>>(t2, stats, n2w, n2b,
                                                             (float*)d_out, 1);
}